// Lovasz_ce_loss_71244917506829
// MI455X (gfx1250) — compile-verified
//
#include <hip/hip_runtime.h>
#include <hip/hip_bf16.h>

#define NBINS 4096
#define HW (1024 * 1024)

typedef float v2f __attribute__((ext_vector_type(2)));
typedef float v8f __attribute__((ext_vector_type(8)));
typedef float f4  __attribute__((ext_vector_type(4)));
typedef int   i4  __attribute__((ext_vector_type(4)));

// ---------------------------------------------------------------------------
// Kernel 0: zero the global histogram workspace (harness does not re-poison).
// ---------------------------------------------------------------------------
__global__ void zero_ws_kernel(unsigned* __restrict__ gh) {
    int i = blockIdx.x * blockDim.x + threadIdx.x;
    if (i < 2 * NBINS) gh[i] = 0u;
}

// ---------------------------------------------------------------------------
// Kernel 1: streaming pass. For C=2 the per-pixel error is the same for both
// classes: e = sigmoid( label==0 ? (x1-x0) : (x0-x1) ).  Bin by descending e
// (bin 0 = largest error) and count per-label occupancy.  One LDS atomic per
// pixel; block-level LDS histogram flushed once with global atomics.
// The 192 MiB input stream is touched exactly once -> non-temporal loads so
// the one-pass stream does not churn the 192 MB global L2.
// ---------------------------------------------------------------------------
__global__ __launch_bounds__(256) void hist_kernel(const float* __restrict__ outs,
                                                   const int*   __restrict__ labels,
                                                   unsigned*    __restrict__ gh,
                                                   int P) {
    __shared__ unsigned lh[2 * NBINS];   // 32 KB
    for (int i = threadIdx.x; i < 2 * NBINS; i += blockDim.x) lh[i] = 0u;
    __syncthreads();

    const int nquad  = P >> 2;
    const int stride = gridDim.x * blockDim.x;
    for (int q = blockIdx.x * blockDim.x + threadIdx.x; q < nquad; q += stride) {
        const int p  = q << 2;
        const int b  = p >> 20;          // p / HW
        const int hw = p & (HW - 1);
        const f4 x0 = __builtin_nontemporal_load((const f4*)(outs + (size_t)b * 2 * HW + hw));
        const f4 x1 = __builtin_nontemporal_load((const f4*)(outs + (size_t)b * 2 * HW + HW + hw));
        const i4 lb = __builtin_nontemporal_load((const i4*)(labels + p));

#pragma unroll
        for (int j = 0; j < 4; ++j) {
            const int  cls = (lb[j] == 0) ? 0 : 1;
            const float s  = (cls == 0) ? (x0[j] - x1[j]) : (x1[j] - x0[j]);
            const float e  = 1.0f / (1.0f + __expf(s));          // error in (0,1)
            int bin = (int)((1.0f - e) * (float)NBINS);          // descending-e order
            bin = bin < 0 ? 0 : (bin > NBINS - 1 ? NBINS - 1 : bin);
            atomicAdd(&lh[cls * NBINS + bin], 1u);
        }
    }
    __syncthreads();

    for (int i = threadIdx.x; i < 2 * NBINS; i += blockDim.x) {
        const unsigned v = lh[i];
        if (v) atomicAdd(&gh[i], v);
    }
}

// ---------------------------------------------------------------------------
// Kernel 2: single-wave (wave32) scan + loss.  Inclusive prefix over the 4096
// bins for both class counts is computed 256 bins at a time as a 16x16 tile:
//     rowScan = X * U_incl   (U = upper-triangular ones)
// realized as 4 chained V_WMMA_F32_16X16X4_F32 (K-slices of 4), then row
// offsets are fixed up through LDS.  The Lovasz dot product telescopes per
// bin:  contribution = e_bin * (J(F_after,G_after) - J(F_before,G_before)),
// J(F,G) = 1 - (gts - F)/(gts + G).
// ---------------------------------------------------------------------------
__global__ __launch_bounds__(32) void scan_kernel(const unsigned* __restrict__ gh,
                                                  float* __restrict__ out) {
    const unsigned* __restrict__ h0 = gh;            // class-0 occupancy per bin
    const unsigned* __restrict__ h1 = gh + NBINS;    // class-1 occupancy per bin
    const int lane  = threadIdx.x;                   // 0..31
    const int m_    = lane & 15;                     // A-matrix row
    const int khalf = lane >> 4;                     // 0 or 1 (K half / row half)
    const int n_    = lane & 15;                     // B/C/D column

    // ---- totals (gts per class) -------------------------------------------
    unsigned t0 = 0u, t1 = 0u;
    for (int i = lane; i < NBINS; i += 32) { t0 += h0[i]; t1 += h1[i]; }
#pragma unroll
    for (int msk = 16; msk >= 1; msk >>= 1) {
        t0 += __shfl_xor(t0, msk, 32);
        t1 += __shfl_xor(t1, msk, 32);
    }
    const float gts0 = (float)t0, gts1 = (float)t1;

    __shared__ float rowtot0[16], rowtot1[16];
    __shared__ float offs0[17], offs1[17];

    float carry0 = 0.0f, carry1 = 0.0f;
    float loss0 = 0.0f, loss1 = 0.0f;

    for (int tile = 0; tile < NBINS / 256; ++tile) {
        const int base = tile * 256;

        // ---- rowScan = X * U_incl via 4 x WMMA f32 16x16x4 ----------------
        v8f c0 = {};
        v8f c1 = {};
#pragma unroll
        for (int t = 0; t < 4; ++t) {
            v2f a0, a1, bu;
#pragma unroll
            for (int j = 0; j < 2; ++j) {
                const int k   = khalf * 2 + j;                 // K within slice
                const int bin = base + m_ * 16 + t * 4 + k;    // X[m][4t+k]
                a0[j] = (float)h0[bin];
                a1[j] = (float)h1[bin];
                // U_incl[4t+k][n] = (4t+k <= n)
                bu[j] = ((t * 4 + k) <= n_) ? 1.0f : 0.0f;
            }
            c0 = __builtin_amdgcn_wmma_f32_16x16x4_f32(false, a0, false, bu,
                                                       (short)0, c0, false, false);
            c1 = __builtin_amdgcn_wmma_f32_16x16x4_f32(false, a1, false, bu,
                                                       (short)0, c1, false, false);
        }
        // c[v] = inclusive row prefix for element (m = v + 8*khalf, n = n_)

        // ---- row totals live in column n==15 (lanes 15 and 31) ------------
        if (n_ == 15) {
#pragma unroll
            for (int v = 0; v < 8; ++v) {
                rowtot0[v + khalf * 8] = c0[v];
                rowtot1[v + khalf * 8] = c1[v];
            }
        }
        __syncthreads();
        if (lane == 0) {
            float s0 = carry0, s1 = carry1;
            for (int r = 0; r < 16; ++r) {
                offs0[r] = s0; s0 += rowtot0[r];
                offs1[r] = s1; s1 += rowtot1[r];
            }
            offs0[16] = s0;
            offs1[16] = s1;
        }
        __syncthreads();

        // ---- per-bin telescoped Lovasz contributions ----------------------
#pragma unroll
        for (int v = 0; v < 8; ++v) {
            const int m   = v + khalf * 8;
            const int bin = base + m * 16 + n_;
            const float F0a = c0[v] + offs0[m];      // inclusive cumsum of h0
            const float F1a = c1[v] + offs1[m];      // inclusive cumsum of h1
            const float u0  = (float)h0[bin];
            const float u1  = (float)h1[bin];
            if (u0 + u1 > 0.0f) {
                const float e = 1.0f - ((float)bin + 0.5f) / (float)NBINS;
                if (t0 > 0u) {   // class 0: fg = h0, bg = h1
                    const float Ja = 1.0f - (gts0 - F0a) / (gts0 + F1a);
                    const float Jb = 1.0f - (gts0 - (F0a - u0)) / (gts0 + (F1a - u1));
                    loss0 += e * (Ja - Jb);
                }
                if (t1 > 0u) {   // class 1: fg = h1, bg = h0
                    const float Ja = 1.0f - (gts1 - F1a) / (gts1 + F0a);
                    const float Jb = 1.0f - (gts1 - (F1a - u1)) / (gts1 + (F0a - u0));
                    loss1 += e * (Ja - Jb);
                }
            }
        }
        carry0 = offs0[16];
        carry1 = offs1[16];
        __syncthreads();
    }

    // ---- reduce and finalize ----------------------------------------------
#pragma unroll
    for (int msk = 16; msk >= 1; msk >>= 1) {
        loss0 += __shfl_xor(loss0, msk, 32);
        loss1 += __shfl_xor(loss1, msk, 32);
    }
    if (lane == 0) {
        const float present = (float)((t0 > 0u) + (t1 > 0u));
        out[0] = (loss0 + loss1) / fmaxf(present, 1.0f);
    }
}

// ---------------------------------------------------------------------------
extern "C" void kernel_launch(void* const* d_in, const int* in_sizes, int n_in,
                              void* d_out, int out_size, void* d_ws, size_t ws_size,
                              hipStream_t stream) {
    const float* outs   = (const float*)d_in[0];   // [16, 2, 1024, 1024] fp32 logits
    const int*   labels = (const int*)d_in[1];     // [16, 1024, 1024] int
    const int    P      = in_sizes[1];             // 16 * 1024 * 1024
    unsigned*    gh     = (unsigned*)d_ws;         // [2][NBINS] u32
    float*       outp   = (float*)d_out;           // scalar loss

    zero_ws_kernel<<<(2 * NBINS + 255) / 256, 256, 0, stream>>>(gh);
    hist_kernel<<<768, 256, 0, stream>>>(outs, labels, gh, P);
    scan_kernel<<<1, 32, 0, stream>>>(gh, outp);
}